// LIDDetector_54408645705820
// MI455X (gfx1250) — compile-verified
//
#include <hip/hip_runtime.h>
#include <math.h>

// LID detector for MI455X (gfx1250, wave32).
//   feats: (L=4, B=256, C=512, S=64) f32   bank: (L, N=50000, C) f32   k = 20
//   out:   (B, L) f32
//
//   1) mean_kernel : q = feats.mean(-1)
//   2) b2_kernel   : b2 = ||bank||^2 rows (wave32-per-row, shfl_xor reduce)
//   3) dist_kernel : v_wmma_f32_16x16x4_f32 GEMM with double-buffered
//                    global_load_async_to_lds_b128 staging of bank chunks
//   4) topk_kernel : 21 smallest per row -> LID formula

typedef __attribute__((ext_vector_type(2))) float v2f;
typedef __attribute__((ext_vector_type(8))) float v8f;

#define L_DIM 4
#define B_DIM 256
#define C_DIM 512
#define S_DIM 64
#define N_DIM 50000
#define K_SEL 20
#define SEL   (K_SEL + 1)      // 21 smallest kept
#define NT    (N_DIM / 16)     // 3125 n-tiles (exact)
#define CCH   32               // c-chunk (floats) staged per async round
#define NCH   (C_DIM / CCH)    // 16 chunks

// ---------------- 1) q = feats.mean(axis=-1) ----------------
__global__ void __launch_bounds__(256)
mean_kernel(const float* __restrict__ feats, float* __restrict__ q) {
  const int idx = blockIdx.x * 256 + threadIdx.x;        // exactly L*B*C threads
  const float4* p = (const float4*)(feats + (size_t)idx * S_DIM);
  float s = 0.f;
#pragma unroll
  for (int i = 0; i < S_DIM / 4; ++i) {
    float4 v = p[i];
    s += v.x + v.y + v.z + v.w;
  }
  q[idx] = s * (1.0f / (float)S_DIM);
}

// ---------------- 2) b2[l,n] = sum_c bank^2 (one wave32 per row) ----------------
__global__ void __launch_bounds__(256)
b2_kernel(const float* __restrict__ bank, float* __restrict__ b2) {
  const int gid  = blockIdx.x * 256 + threadIdx.x;
  const int row  = gid >> 5;                             // exactly L*N rows
  const int lane = gid & 31;
  const float* r = bank + (size_t)row * C_DIM;
  float s = 0.f;
  for (int c = lane; c < C_DIM; c += 32) { float v = r[c]; s += v * v; }
#pragma unroll
  for (int off = 16; off > 0; off >>= 1) s += __shfl_xor(s, off, 32);
  if (lane == 0) b2[row] = s;
}

// ---------------- 3) distances: WMMA f32 + async-to-LDS double buffering ----------------
// grid.x = L*(B/16) = 64 ; grid.y = 16 segments ; block = 128 (4 waves)
__global__ void __launch_bounds__(128)
dist_kernel(const float* __restrict__ q, const float* __restrict__ bank,
            const float* __restrict__ b2, float* __restrict__ dist) {
  __shared__ float qs[16 * C_DIM];                       // 32 KB q-tile
  __shared__ float q2s[16];
  __shared__ __align__(16) float bbuf[4][2][16 * CCH];   // 16 KB: per-wave double buffers

  const int l  = blockIdx.x >> 4;
  const int m0 = (blockIdx.x & 15) << 4;

  const float* qrow = q + (size_t)(l * B_DIM + m0) * C_DIM;
  for (int i = threadIdx.x; i < (16 * C_DIM) / 4; i += 128)
    ((float4*)qs)[i] = ((const float4*)qrow)[i];
  __syncthreads();
  if (threadIdx.x < 16) {
    float s = 0.f;
    const float* r = qs + threadIdx.x * C_DIM;
    for (int c = 0; c < C_DIM; ++c) { float v = r[c]; s += v * v; }
    q2s[threadIdx.x] = s;
  }
  __syncthreads();

  const int lane  = threadIdx.x & 31;
  const int wave  = threadIdx.x >> 5;
  const int mf    = lane & 15;            // A: row M ; B/C/D: col N
  const int kbase = (lane >> 4) << 1;     // lanes 16-31 hold K=2,3 (ISA 7.12.2)
  const int mrow  = lane >> 4;            // C/D: M = r + 8*mrow

  // async staging geometry: instr i, lane -> row 4i + (lane>>3), col (lane&7)*4
  const int n_l = lane >> 3;              // 0..3
  const int c_l = (lane & 7) * 4;         // 0..28 (floats), 16B aligned

  // LDS byte offsets of this wave's two buffers (generic ptr low 32 bits == LDS addr)
  const unsigned lb0 = (unsigned)(size_t)(&bbuf[wave][0][0]);
  const unsigned lb1 = (unsigned)(size_t)(&bbuf[wave][1][0]);

  const float* bl   = bank + (size_t)l * N_DIM * C_DIM;  // SGPR base for GVS async
  const float* b2l  = b2 + (size_t)l * N_DIM;
  float*       drow = dist + (size_t)(l * B_DIM + m0) * N_DIM;

  const int worker   = blockIdx.y * 4 + wave;
  const int nworkers = gridDim.y * 4;

  const float* apf = qs + mf * C_DIM + kbase;

  for (int t = worker; t < NT; t += nworkers) {   // uniform per wave: EXEC all-ones
    const int n0 = t << 4;
    if (t + nworkers < NT)                        // global_prefetch_b8 next tile
      __builtin_prefetch(bl + (size_t)(((t + nworkers) << 4) + n_l) * C_DIM, 0, 0);

    // per-lane fixed parts of the async addresses for this n-tile
    const int gbase = (n0 + n_l) * (C_DIM * 4) + c_l * 4;      // byte off vs bl
    const unsigned lsub = (unsigned)(n_l * CCH + c_l) * 4u;    // byte off in buffer

    // issue one 16x32 chunk (4 x b128-per-lane async transfers)
#define ISSUE_CHUNK(ch, ldsbase)                                               \
    {                                                                          \
      _Pragma("unroll")                                                        \
      for (int i = 0; i < 4; ++i) {                                            \
        unsigned laddr = (ldsbase) + lsub + (unsigned)(i * 4 * CCH * 4);       \
        int      goff  = gbase + (ch) * (CCH * 4) + i * 4 * (C_DIM * 4);       \
        asm volatile("global_load_async_to_lds_b128 %0, %1, %2"                \
                     :: "v"(laddr), "v"(goff), "s"(bl) : "memory");            \
      }                                                                        \
    }

    v8f acc = {};
    ISSUE_CHUNK(0, lb0)
    for (int ch = 0; ch < NCH; ++ch) {
      asm volatile("s_wait_asynccnt 0x0" ::: "memory");        // chunk ch landed
      if (ch + 1 < NCH)                                        // overlap next fetch
        ISSUE_CHUNK(ch + 1, ((ch & 1) ? lb0 : lb1))

      const float* bchunk = &bbuf[wave][ch & 1][0];
      const float2* bp = (const float2*)(bchunk + mf * CCH + kbase);
      const float2* ac = (const float2*)(apf + ch * CCH);
#pragma unroll
      for (int kk = 0; kk < CCH; kk += 4) {
        float2 af = ac[kk >> 1];                               // ds_load_b64 (q-tile)
        float2 bf = bp[kk >> 1];                               // ds_load_b64 (staged)
        v2f a; a.x = af.x; a.y = af.y;
        v2f b; b.x = bf.x; b.y = bf.y;
        acc = __builtin_amdgcn_wmma_f32_16x16x4_f32(
            false, a, false, b, (short)0, acc, false, false);
      }
    }
#undef ISSUE_CHUNK

    const float b2v = b2l[n0 + mf];
#pragma unroll
    for (int r = 0; r < 8; ++r) {
      const int m   = r + (mrow << 3);
      const float d2 = q2s[m] + b2v - 2.0f * acc[r];
      drow[(size_t)m * N_DIM + n0 + mf] = sqrtf(fmaxf(d2, 0.0f));
    }
  }
}

// ---------------- 4) top-(k+1) per row + LID ----------------
__global__ void __launch_bounds__(256)
topk_kernel(const float* __restrict__ dist, float* __restrict__ out) {
  const int row = blockIdx.x;                      // l*B + b
  const int l   = row >> 8;
  const int b   = row & 255;
  const int tid = threadIdx.x;
  const float* dr = dist + (size_t)row * N_DIM;
  const float BIG = 3.0e38f;

  float loc[SEL];                                  // sorted ascending, static idx only
#pragma unroll
  for (int i = 0; i < SEL; ++i) loc[i] = BIG;
  for (int i = tid; i < N_DIM; i += 256) {
    float v = dr[i];
    if (v < loc[SEL - 1]) {
      loc[SEL - 1] = v;
#pragma unroll
      for (int j = SEL - 1; j > 0; --j)
        if (loc[j] < loc[j - 1]) { float t = loc[j]; loc[j] = loc[j - 1]; loc[j - 1] = t; }
    }
  }

  __shared__ float cand[256 * SEL];
  __shared__ float minv[256];
  __shared__ int   mini[256];
  __shared__ float sel[SEL];
#pragma unroll
  for (int i = 0; i < SEL; ++i) cand[tid * SEL + i] = loc[i];
  __syncthreads();

  for (int r = 0; r < SEL; ++r) {
    float mv = BIG; int mi = tid * SEL;
#pragma unroll
    for (int i = 0; i < SEL; ++i) {
      float c = cand[tid * SEL + i];
      if (c < mv) { mv = c; mi = tid * SEL + i; }
    }
    minv[tid] = mv; mini[tid] = mi;
    __syncthreads();
    for (int s = 128; s > 0; s >>= 1) {
      if (tid < s && minv[tid + s] < minv[tid]) { minv[tid] = minv[tid + s]; mini[tid] = mini[tid + s]; }
      __syncthreads();
    }
    if (tid == 0) { sel[r] = minv[0]; cand[mini[0]] = BIG; }
    __syncthreads();
  }

  if (tid == 0) {
    const float rk = sel[SEL - 1];                 // k-th neighbor (excl. nearest)
    float s = 0.f;
#pragma unroll
    for (int i = 1; i < SEL; ++i) s += logf(sel[i] / rk);
    out[b * L_DIM + l] = -((float)K_SEL / s);      // lids, transposed to (B, L)
  }
}

// ---------------- launch ----------------
extern "C" void kernel_launch(void* const* d_in, const int* in_sizes, int n_in,
                              void* d_out, int out_size, void* d_ws, size_t ws_size,
                              hipStream_t stream) {
  (void)in_sizes; (void)n_in; (void)out_size; (void)ws_size;
  const float* feats = (const float*)d_in[0];
  const float* bank  = (const float*)d_in[1];
  // d_in[2] is k==20, baked in at compile time.
  float* out = (float*)d_out;

  float* q    = (float*)d_ws;                                  //  2.0 MB
  float* b2   = q  + (size_t)L_DIM * B_DIM * C_DIM;            //  0.8 MB
  float* dist = b2 + (size_t)L_DIM * N_DIM;                    // 204.8 MB

  mean_kernel<<<(L_DIM * B_DIM * C_DIM) / 256, 256, 0, stream>>>(feats, q);
  b2_kernel  <<<(L_DIM * N_DIM) / 8,        256, 0, stream>>>(bank, b2);
  dim3 g(L_DIM * (B_DIM / 16), 16);
  dist_kernel<<<g, 128, 0, stream>>>(q, bank, b2, dist);
  topk_kernel<<<L_DIM * B_DIM, 256, 0, stream>>>(dist, out);
}